// ScatterModel_64690797413098
// MI455X (gfx1250) — compile-verified
//
#include <hip/hip_runtime.h>

// Scatter-add: out[index[e], :] += src[e, :]
// N_EDGES = 1.6M, N_NODES = 50K, D_FEAT = 32 (f32).
// Memory-bound (~218 MB total traffic, ~9.4 us at 23.3 TB/s HBM).
// Strategy: 128-bit NT streaming loads of src, hardware f32 atomics into
// L2-resident out (6.4 MB << 192 MB L2).
//
// NOTE: scatter_add_f32 is defined FIRST so its disassembly appears in the
// compile-feedback snippet (verifying global_atomic_add_f32 vs CAS loop).

typedef __attribute__((ext_vector_type(4))) float v4f;

constexpr int D_FEAT = 32;
constexpr int VEC = 4;                         // float4 per thread
constexpr int GROUPS_PER_EDGE = D_FEAT / VEC;  // 8 lanes cover one edge row

__global__ __launch_bounds__(256) void scatter_add_f32(
    const float* __restrict__ src, const int* __restrict__ index,
    float* __restrict__ out, int n_groups) {
  int g = blockIdx.x * blockDim.x + threadIdx.x;
  if (g >= n_groups) return;

  int e   = g >> 3;  // edge id        (n_groups = n_edges * 8)
  int sub = g & 7;   // float4 chunk within the 32-float row

  // 8 consecutive lanes read the same index dword; coalescer merges. The
  // index stream is read-effectively-once (6.4 MB) -> non-temporal.
  int node = __builtin_nontemporal_load(index + e);

  // Streaming read of the src row chunk: read-once -> non-temporal so the
  // 205 MB stream does not push the out array out of L2.
  const v4f* sp = (const v4f*)src + g;
  v4f v = __builtin_nontemporal_load(sp);

  // Atomic combine in L2 (out fits entirely in the 192 MB L2). Result is
  // discarded -> non-returning global_atomic_add_f32 expected.
  float* op = out + (size_t)node * D_FEAT + (size_t)sub * VEC;
  __hip_atomic_fetch_add(op + 0, v.x, __ATOMIC_RELAXED, __HIP_MEMORY_SCOPE_AGENT);
  __hip_atomic_fetch_add(op + 1, v.y, __ATOMIC_RELAXED, __HIP_MEMORY_SCOPE_AGENT);
  __hip_atomic_fetch_add(op + 2, v.z, __ATOMIC_RELAXED, __HIP_MEMORY_SCOPE_AGENT);
  __hip_atomic_fetch_add(op + 3, v.w, __ATOMIC_RELAXED, __HIP_MEMORY_SCOPE_AGENT);
}

__global__ __launch_bounds__(256) void scatter_init_out(
    const float* __restrict__ out_init, float* __restrict__ out, int n4) {
  int i = blockIdx.x * blockDim.x + threadIdx.x;
  if (i < n4) {
    const v4f* s = (const v4f*)out_init;
    v4f* d = (v4f*)out;
    d[i] = s[i];
  }
}

extern "C" void kernel_launch(void* const* d_in, const int* in_sizes, int n_in,
                              void* d_out, int out_size, void* d_ws, size_t ws_size,
                              hipStream_t stream) {
  const float* src      = (const float*)d_in[0];  // [n_edges, 32] f32
  const int*   index    = (const int*)d_in[1];    // [n_edges]     i32
  const float* out_init = (const float*)d_in[2];  // [n_nodes, 32] f32 (zeros)
  float*       out      = (float*)d_out;          // [n_nodes, 32] f32

  const int n_src = in_sizes[0];  // n_edges * 32
  const int n_out = out_size;     // n_nodes * 32

  // 1) Initialize out from the provided out input (d_out is poisoned).
  {
    int n4 = n_out / 4;
    int blocks = (n4 + 255) / 256;
    scatter_init_out<<<blocks, 256, 0, stream>>>(out_init, out, n4);
  }

  // 2) Scatter-add, one float4 per thread.
  {
    int n_groups = n_src / 4;  // = n_edges * GROUPS_PER_EDGE
    int blocks = (n_groups + 255) / 256;
    scatter_add_f32<<<blocks, 256, 0, stream>>>(src, index, out, n_groups);
  }
  (void)GROUPS_PER_EDGE; (void)d_ws; (void)ws_size; (void)n_in;
}